// TFF_Dihedral_76020921139247
// MI455X (gfx1250) — compile-verified
//
#include <hip/hip_runtime.h>
#include <hip/hip_bf16.h>

#define NATOMS 4096
#define NF (NATOMS * 3)          // 12288 force components
#define BLK 256
#define NBLOCKS 1024

typedef __attribute__((ext_vector_type(2))) float v2f;
typedef __attribute__((ext_vector_type(4))) float v4f;
typedef __attribute__((ext_vector_type(8))) float v8f;
typedef __attribute__((ext_vector_type(4))) int   v4i;

struct F3 { float x, y, z; };

__device__ __forceinline__ F3 cross3(F3 a, F3 b) {
    return { a.y * b.z - a.z * b.y,
             a.z * b.x - a.x * b.z,
             a.x * b.y - a.y * b.x };
}
__device__ __forceinline__ float dot3(F3 a, F3 b) {
    return a.x * b.x + a.y * b.y + a.z * b.z;
}

// r_ab = vector_mat[a,b] * (-dist_mat[a,b])
__device__ __forceinline__ F3 pairvec(const float* __restrict__ dist,
                                      const float* __restrict__ vec,
                                      long p) {
    float d = -dist[p];
    const float* v = vec + 3 * p;
    return { v[0] * d, v[1] * d, v[2] * d };
}

__global__ void TFF_Dihedral_init(const float* __restrict__ forces_in,
                                  float* __restrict__ out) {
    int i = blockIdx.x * blockDim.x + threadIdx.x;
    if (i == 0) out[0] = 0.0f;                 // energy accumulator
    if (i < NF) out[1 + i] = forces_in[i];     // forces base = forces_out input
}

__global__ __launch_bounds__(BLK) void TFF_Dihedral_main(
    const float* __restrict__ dist,      // [4096,4096]
    const float* __restrict__ vec,       // [4096,4096,3]
    const float* __restrict__ params,    // [N,4,3] (per,k0,phi0)
    const int*   __restrict__ cidx,      // [N,4]
    float* __restrict__ out,             // [1+12288]
    int ndih)
{
    __shared__ float ldsF[NF];           // 48 KB per-block force accumulator
    __shared__ float ldsE;

    for (int j = threadIdx.x; j < NF; j += BLK) ldsF[j] = 0.0f;
    if (threadIdx.x == 0) ldsE = 0.0f;
    __syncthreads();

    const int stride = gridDim.x * BLK;
    int i = blockIdx.x * BLK + threadIdx.x;
    float e_acc = 0.0f;

    const v4i* cidx4 = reinterpret_cast<const v4i*>(cidx);
    const v4f* prm4  = reinterpret_cast<const v4f*>(params);

    v4i idx = {0, 0, 0, 0};
    // Streaming, read-once data: non-temporal loads keep the 192MB L2 free
    // for the 256MB random-gather working set (vector_mat/dist_mat).
    if (i < ndih) idx = __builtin_nontemporal_load(cidx4 + i);

    while (i < ndih) {
        // ---- software pipeline: fetch next idx, prefetch its gather lines ----
        int inext = i + stride;
        v4i idxn = idx;
        if (inext < ndih) {
            idxn = __builtin_nontemporal_load(cidx4 + inext);
            long q1 = (long)idxn.x * NATOMS + idxn.y;
            long q2 = (long)idxn.y * NATOMS + idxn.z;
            long q3 = (long)idxn.z * NATOMS + idxn.w;
            __builtin_prefetch(dist + q1, 0, 3);   // global_prefetch_b8
            __builtin_prefetch(dist + q2, 0, 3);
            __builtin_prefetch(dist + q3, 0, 3);
            __builtin_prefetch(vec + 3 * q1, 0, 3);
            __builtin_prefetch(vec + 3 * q2, 0, 3);
            __builtin_prefetch(vec + 3 * q3, 0, 3);
        }

        // ---- gathers for current dihedral (RT hint: benefits from L2 reuse) ----
        long p1 = (long)idx.x * NATOMS + idx.y;
        long p2 = (long)idx.y * NATOMS + idx.z;
        long p3 = (long)idx.z * NATOMS + idx.w;
        F3 r12 = pairvec(dist, vec, p1);
        F3 r23 = pairvec(dist, vec, p2);
        F3 r34 = pairvec(dist, vec, p3);

        F3 cA = cross3(r12, r23);
        F3 cB = cross3(r23, r34);
        F3 cC = cross3(r23, cA);

        float nA2 = dot3(cA, cA);
        float nB2 = dot3(cB, cB);
        float nC2 = dot3(cC, cC);
        float nA = sqrtf(nA2), nB = sqrtf(nB2), nC = sqrtf(nC2);

        float cosPhi = dot3(cA, cB) / (nA * nB);
        float sinPhi = dot3(cC, cB) / (nC * nB);
        float phi = -atan2f(sinPhi, cosPhi);

        // ---- 4 torsion terms: 3x non-temporal b128 param loads ----
        const v4f* pbase = prm4 + (long)i * 3;
        v4f q0 = __builtin_nontemporal_load(pbase + 0);
        v4f q1 = __builtin_nontemporal_load(pbase + 1);
        v4f q2 = __builtin_nontemporal_load(pbase + 2);
        float prm[12] = { q0.x, q0.y, q0.z, q0.w,
                          q1.x, q1.y, q1.z, q1.w,
                          q2.x, q2.y, q2.z, q2.w };

        float coeff = 0.0f;
        #pragma unroll
        for (int t = 0; t < 4; ++t) {
            float per = prm[3 * t + 0];
            float k0  = prm[3 * t + 1];
            float ph0 = prm[3 * t + 2];
            float ang = per * phi - ph0;
            float sa, ca;
            sincosf(ang, &sa, &ca);
            e_acc += k0 * (1.0f + ca);
            coeff -= per * k0 * sa;
        }

        float n2d2 = dot3(r23, r23);
        float nd2  = sqrtf(n2d2);
        float ff0 = -coeff * nd2 / nA2;
        float ff1 = dot3(r12, r23) / n2d2;
        float ff2 = dot3(r34, r23) / n2d2;
        float ff3 =  coeff * nd2 / nB2;

        F3 f0v = { ff0 * cA.x, ff0 * cA.y, ff0 * cA.z };
        F3 f3v = { ff3 * cB.x, ff3 * cB.y, ff3 * cB.z };
        F3 s   = { ff1 * f0v.x - ff2 * f3v.x,
                   ff1 * f0v.y - ff2 * f3v.y,
                   ff1 * f0v.z - ff2 * f3v.z };

        // ---- scatter into LDS (ds_add_f32), 12 adds ----
        atomicAdd(&ldsF[3 * idx.x + 0], -f0v.x);
        atomicAdd(&ldsF[3 * idx.x + 1], -f0v.y);
        atomicAdd(&ldsF[3 * idx.x + 2], -f0v.z);
        atomicAdd(&ldsF[3 * idx.y + 0],  f0v.x + s.x);
        atomicAdd(&ldsF[3 * idx.y + 1],  f0v.y + s.y);
        atomicAdd(&ldsF[3 * idx.y + 2],  f0v.z + s.z);
        atomicAdd(&ldsF[3 * idx.z + 0],  f3v.x - s.x);
        atomicAdd(&ldsF[3 * idx.z + 1],  f3v.y - s.y);
        atomicAdd(&ldsF[3 * idx.z + 2],  f3v.z - s.z);
        atomicAdd(&ldsF[3 * idx.w + 0], -f3v.x);
        atomicAdd(&ldsF[3 * idx.w + 1], -f3v.y);
        atomicAdd(&ldsF[3 * idx.w + 2], -f3v.z);

        idx = idxn;
        i = inext;
    }

    // ---- per-wave energy reduction on the matrix pipe ----
    // A(16x4) = per-lane partials (VGPR0=e, VGPR1=0), B(4x16) = ones.
    // D row m = e(m) + e(m+16), replicated across columns; summing the 8 D
    // VGPRs gives half-wave totals; one xor-16 shuffle completes the wave sum.
#if __has_builtin(__builtin_amdgcn_wmma_f32_16x16x4_f32)
    {
        v2f a; a.x = e_acc; a.y = 0.0f;
        v2f b; b.x = 1.0f;  b.y = 1.0f;
        v8f c = {};
        c = __builtin_amdgcn_wmma_f32_16x16x4_f32(
                /*neg_a=*/false, a, /*neg_b=*/false, b,
                /*c_mod=*/(short)0, c, /*reuse_a=*/false, /*reuse_b=*/false);
        float wsum = c[0] + c[1] + c[2] + c[3] + c[4] + c[5] + c[6] + c[7];
        wsum += __shfl_xor(wsum, 16, 32);
        if ((threadIdx.x & 31) == 0) atomicAdd(&ldsE, wsum);
    }
#else
    {
        float wsum = e_acc;
        for (int m = 16; m >= 1; m >>= 1) wsum += __shfl_xor(wsum, m, 32);
        if ((threadIdx.x & 31) == 0) atomicAdd(&ldsE, wsum);
    }
#endif
    __syncthreads();

    // ---- flush block accumulators to global ----
    if (threadIdx.x == 0) atomicAdd(&out[0], ldsE);
    for (int j = threadIdx.x; j < NF; j += BLK) {
        float v = ldsF[j];
        if (v != 0.0f) atomicAdd(&out[1 + j], v);
    }
}

extern "C" void kernel_launch(void* const* d_in, const int* in_sizes, int n_in,
                              void* d_out, int out_size, void* d_ws, size_t ws_size,
                              hipStream_t stream) {
    const float* dist   = (const float*)d_in[0];   // dist_mat   [4096,4096]
    const float* vec    = (const float*)d_in[1];   // vector_mat [4096,4096,3]
    const float* fbase  = (const float*)d_in[2];   // forces_out [4096,3]
    const float* params = (const float*)d_in[3];   // params     [N,4,3]
    const int*   cidx   = (const int*)d_in[4];     // coord_idx  [N,4]
    float* out = (float*)d_out;                    // [0]=energy, [1..]=forces
    int ndih = in_sizes[4] / 4;

    TFF_Dihedral_init<<<(NF + BLK - 1) / BLK, BLK, 0, stream>>>(fbase, out);
    TFF_Dihedral_main<<<NBLOCKS, BLK, 0, stream>>>(dist, vec, params, cidx, out, ndih);
}